// TopoGNN_76373108457891
// MI455X (gfx1250) — compile-verified
//
#include <hip/hip_runtime.h>
#include <hip/hip_bf16.h>

// ---------------------------------------------------------------------------
// EdgeGAT x2 + linear head for MI455X (gfx1250, wave32).
// Sparse passes: factorized per-edge attention with uint-key atomicMax softmax.
// Dense GEMMs (x1@W2, cat@Wef): V_WMMA_F32_16X16X4_F32, fp32 end to end.
// WMMA kernels: clamped gathers (store-guarded), b64 A loads, preloaded
// fragments -> one load clause + 4 back-to-back v_wmma per tile.
// ---------------------------------------------------------------------------

typedef __attribute__((ext_vector_type(2))) float v2f;
typedef __attribute__((ext_vector_type(8))) float v8f;

#define NN 50000
#define NEG_SLOPE 0.2f

// Monotone float -> uint key for atomicMax-based segment max (handles negatives).
__device__ __forceinline__ unsigned fkey(float f) {
  unsigned u = __float_as_uint(f);
  return (int)u >= 0 ? (u | 0x80000000u) : ~u;
}
__device__ __forceinline__ float fdekey(unsigned k) {
  unsigned u = (k & 0x80000000u) ? (k & 0x7FFFFFFFu) : ~k;
  return __uint_as_float(u);
}

__device__ __forceinline__ void edge_decode(int e, int E_, const int* __restrict__ src,
                                            const int* __restrict__ dst,
                                            const float* __restrict__ ef,
                                            int& s, int& d, float& ev) {
  if (e < E_) { s = src[e]; d = dst[e]; ev = ef[e]; }
  else        { s = d = e - E_; ev = 0.f; }          // self-loop, efeature 0
}

// --- coefficient precompute + padded Wef ------------------------------------
// coef layout: [0:16) cal1, [16:32) cer1, [32:48) cee1, [48:64) b1col,
//              [64:72) cee2, [72:80) b2col, [128:384) WefP (16x16, cols 8..15 = 0)
__global__ void k_prep(const float* __restrict__ W1, const float* __restrict__ We1,
                       const float* __restrict__ al1, const float* __restrict__ ar1,
                       const float* __restrict__ ae1, const float* __restrict__ b1,
                       const float* __restrict__ We2, const float* __restrict__ ae2,
                       const float* __restrict__ b2, const float* __restrict__ Wef,
                       float* __restrict__ coef) {
  int t = threadIdx.x;
  if (t < 16) {
    float ca = 0.f, cr = 0.f, ce = 0.f, bc = 0.f;
    for (int f = 0; f < 16; ++f) {
      float w = W1[t * 16 + f];
      ca += w * al1[t * 16 + f];
      cr += w * ar1[t * 16 + f];
      ce += We1[t * 16 + f] * ae1[t * 16 + f];
      bc += b1[f * 16 + t];                // sum over heads of column t
    }
    coef[t] = ca; coef[16 + t] = cr; coef[32 + t] = ce; coef[48 + t] = bc;
    if (t < 8) {
      float c2 = 0.f, b2c = 0.f;
      for (int f = 0; f < 8; ++f) {
        c2  += We2[t * 8 + f] * ae2[t * 8 + f];
        b2c += b2[f * 8 + t];
      }
      coef[64 + t] = c2; coef[72 + t] = b2c;
    }
  }
  if (t < 256) {                           // zero-padded 16x16 Wef
    int row = t >> 4, col = t & 15;
    coef[128 + t] = (col < 8) ? Wef[row * 8 + col] : 0.f;
  }
}

// --- init layer-1 scratch ---------------------------------------------------
__global__ void k_init1(float* __restrict__ nfeat, unsigned* __restrict__ m1,
                        float* __restrict__ den1, float* __restrict__ P1,
                        float* __restrict__ Q1, int N16) {
  int i = blockIdx.x * blockDim.x + threadIdx.x;
  if (i < N16) {
    m1[i] = 0u; den1[i] = 0.f; P1[i] = 0.f; Q1[i] = 0.f;
    if (i < NN) nfeat[i] = 0.f;
  }
}

// nfeat[n] = sum of incoming capacities (self-loop capacity is 0)
__global__ void k_nfeat(const int* __restrict__ dst, const float* __restrict__ cap,
                        float* __restrict__ nfeat, int E_) {
  int e = blockIdx.x * blockDim.x + threadIdx.x;
  if (e < E_) atomicAdd(&nfeat[dst[e]], cap[e]);
}

// --- layer-1 edge softmax passes (s recomputed each pass; it's 3 FMAs/head) -
__global__ void k_passA1(const int* __restrict__ src, const int* __restrict__ dst,
                         const float* __restrict__ ef, const float* __restrict__ nfeat,
                         const float* __restrict__ coef, unsigned* __restrict__ m1,
                         int E_, int Et) {
  int e = blockIdx.x * blockDim.x + threadIdx.x;
  if (e >= Et) return;
  int s, d; float ev; edge_decode(e, E_, src, dst, ef, s, d, ev);
  float ns = nfeat[s], nd = nfeat[d];
#pragma unroll
  for (int h = 0; h < 16; ++h) {
    float v = ns * coef[h] + nd * coef[16 + h] + ev * coef[32 + h];
    v = v > 0.f ? v : NEG_SLOPE * v;
    atomicMax(&m1[d * 16 + h], fkey(v));
  }
}

__global__ void k_passB1(const int* __restrict__ src, const int* __restrict__ dst,
                         const float* __restrict__ ef, const float* __restrict__ nfeat,
                         const float* __restrict__ coef, const unsigned* __restrict__ m1,
                         float* __restrict__ den1, int E_, int Et) {
  int e = blockIdx.x * blockDim.x + threadIdx.x;
  if (e >= Et) return;
  int s, d; float ev; edge_decode(e, E_, src, dst, ef, s, d, ev);
  float ns = nfeat[s], nd = nfeat[d];
#pragma unroll
  for (int h = 0; h < 16; ++h) {
    float v = ns * coef[h] + nd * coef[16 + h] + ev * coef[32 + h];
    v = v > 0.f ? v : NEG_SLOPE * v;
    atomicAdd(&den1[d * 16 + h], __expf(v - fdekey(m1[d * 16 + h])));
  }
}

__global__ void k_passC1(const int* __restrict__ src, const int* __restrict__ dst,
                         const float* __restrict__ ef, const float* __restrict__ nfeat,
                         const float* __restrict__ coef, const unsigned* __restrict__ m1,
                         const float* __restrict__ den1, float* __restrict__ P1,
                         float* __restrict__ Q1, int E_, int Et) {
  int e = blockIdx.x * blockDim.x + threadIdx.x;
  if (e >= Et) return;
  int s, d; float ev; edge_decode(e, E_, src, dst, ef, s, d, ev);
  float ns = nfeat[s], nd = nfeat[d];
#pragma unroll
  for (int h = 0; h < 16; ++h) {
    float v = ns * coef[h] + nd * coef[16 + h] + ev * coef[32 + h];
    v = v > 0.f ? v : NEG_SLOPE * v;
    float ex = __expf(v - fdekey(m1[d * 16 + h]));
    float a = ex / den1[d * 16 + h];
    atomicAdd(&P1[d * 16 + h], a * ns);
    atomicAdd(&Q1[d * 16 + h], a * ev);
  }
}

// x1[n,f] = relu( sum_h W1[h,f]*P1[n,h] + We1[h,f]*Q1[n,h] + b1col[f] )
__global__ void k_x1(const float* __restrict__ W1, const float* __restrict__ We1,
                     const float* __restrict__ P1, const float* __restrict__ Q1,
                     const float* __restrict__ coef, float* __restrict__ x1, int N16) {
  int i = blockIdx.x * blockDim.x + threadIdx.x;
  if (i >= N16) return;
  int n = i >> 4, f = i & 15;
  float acc = coef[48 + f];
#pragma unroll
  for (int h = 0; h < 16; ++h)
    acc += W1[h * 16 + f] * P1[n * 16 + h] + We1[h * 16 + f] * Q1[n * 16 + h];
  x1[i] = acc > 0.f ? acc : 0.f;
}

// --- feat2 = x1(N x 16) @ W2(16 x 64) via V_WMMA_F32_16X16X4_F32 ------------
// one wave per (16-node tile, 16-col tile); preload frags, 4 chained WMMAs
__global__ void k_feat2_wmma(const float* __restrict__ x1, const float* __restrict__ W2,
                             float* __restrict__ feat2, int nWaves) {
  int wave = (blockIdx.x * blockDim.x + threadIdx.x) >> 5;
  int lane = threadIdx.x & 31;
  if (wave >= nWaves) return;            // wave-uniform: EXEC stays all-ones
  int mt = wave >> 2;                    // node tile
  int ct = wave & 3;                     // column tile (16 cols each)
  int half = lane >> 4;                  // K-half select (ISA 16x4 layout)
  int m = lane & 15;                     // A row within tile / B,N column
  int node = mt * 16 + m;
  int nodeC = node < NN ? node : NN - 1; // clamp: garbage rows are store-guarded
  v2f af[4], bf[4];
#pragma unroll
  for (int kk = 0; kk < 4; ++kk) {
    int k0 = kk * 4 + 2 * half;          // k0 even -> aligned float2
    af[kk] = *(const v2f*)(x1 + nodeC * 16 + k0);
    bf[kk].x = W2[k0 * 64 + ct * 16 + m];
    bf[kk].y = W2[(k0 + 1) * 64 + ct * 16 + m];
  }
  v8f acc = {};
#pragma unroll
  for (int kk = 0; kk < 4; ++kk)
    acc = __builtin_amdgcn_wmma_f32_16x16x4_f32(false, af[kk], false, bf[kk],
                                                (short)0, acc, false, false);
#pragma unroll
  for (int v = 0; v < 8; ++v) {
    int row = mt * 16 + v + 8 * half;    // C/D layout: M = vgpr + 8*(lane>=16)
    if (row < NN) feat2[row * 64 + ct * 16 + m] = acc[v];
  }
}

// el2/er2: per-(node,head) contractions of feat2 with al2/ar2
__global__ void k_el2er2(const float* __restrict__ feat2, const float* __restrict__ al2,
                         const float* __restrict__ ar2, float* __restrict__ el2,
                         float* __restrict__ er2, int N8) {
  int i = blockIdx.x * blockDim.x + threadIdx.x;
  if (i >= N8) return;
  int n = i >> 3, h = i & 7;
  float a = 0.f, b = 0.f;
#pragma unroll
  for (int f = 0; f < 8; ++f) {
    float fv = feat2[n * 64 + h * 8 + f];
    a += fv * al2[h * 8 + f];
    b += fv * ar2[h * 8 + f];
  }
  el2[i] = a; er2[i] = b;
}

__global__ void k_init2(unsigned* __restrict__ m2, float* __restrict__ den2,
                        float* __restrict__ x2, const float* __restrict__ coef, int N8) {
  int i = blockIdx.x * blockDim.x + threadIdx.x;
  if (i < N8) { m2[i] = 0u; den2[i] = 0.f; x2[i] = coef[72 + (i & 7)]; }
}

// --- layer-2 edge softmax passes --------------------------------------------
__global__ void k_passA2(const int* __restrict__ src, const int* __restrict__ dst,
                         const float* __restrict__ ef, const float* __restrict__ el2,
                         const float* __restrict__ er2, const float* __restrict__ coef,
                         unsigned* __restrict__ m2, int E_, int Et) {
  int e = blockIdx.x * blockDim.x + threadIdx.x;
  if (e >= Et) return;
  int s, d; float ev; edge_decode(e, E_, src, dst, ef, s, d, ev);
#pragma unroll
  for (int h = 0; h < 8; ++h) {
    float v = el2[s * 8 + h] + er2[d * 8 + h] + ev * coef[64 + h];
    v = v > 0.f ? v : NEG_SLOPE * v;
    atomicMax(&m2[d * 8 + h], fkey(v));
  }
}

__global__ void k_passB2(const int* __restrict__ src, const int* __restrict__ dst,
                         const float* __restrict__ ef, const float* __restrict__ el2,
                         const float* __restrict__ er2, const float* __restrict__ coef,
                         const unsigned* __restrict__ m2, float* __restrict__ den2,
                         int E_, int Et) {
  int e = blockIdx.x * blockDim.x + threadIdx.x;
  if (e >= Et) return;
  int s, d; float ev; edge_decode(e, E_, src, dst, ef, s, d, ev);
#pragma unroll
  for (int h = 0; h < 8; ++h) {
    float v = el2[s * 8 + h] + er2[d * 8 + h] + ev * coef[64 + h];
    v = v > 0.f ? v : NEG_SLOPE * v;
    atomicAdd(&den2[d * 8 + h], __expf(v - fdekey(m2[d * 8 + h])));
  }
}

// pass C2 folds the head-sum: x2[d,f] += sum_h a[h]*(feat2[s,h,f] + e*We2[h,f])
__global__ void k_passC2(const int* __restrict__ src, const int* __restrict__ dst,
                         const float* __restrict__ ef, const float* __restrict__ el2,
                         const float* __restrict__ er2, const float* __restrict__ coef,
                         const unsigned* __restrict__ m2, const float* __restrict__ den2,
                         const float* __restrict__ feat2, const float* __restrict__ We2,
                         float* __restrict__ x2, int E_, int Et) {
  int e = blockIdx.x * blockDim.x + threadIdx.x;
  if (e >= Et) return;
  int s, d; float ev; edge_decode(e, E_, src, dst, ef, s, d, ev);
  float ah[8];
#pragma unroll
  for (int h = 0; h < 8; ++h) {
    float v = el2[s * 8 + h] + er2[d * 8 + h] + ev * coef[64 + h];
    v = v > 0.f ? v : NEG_SLOPE * v;
    float ex = __expf(v - fdekey(m2[d * 8 + h]));
    ah[h] = ex / den2[d * 8 + h];
  }
#pragma unroll
  for (int f = 0; f < 8; ++f) {
    float val = 0.f;
#pragma unroll
    for (int h = 0; h < 8; ++h)
      val += ah[h] * (feat2[s * 64 + h * 8 + f] + ev * We2[h * 8 + f]);
    atomicAdd(&x2[d * 8 + f], val);
  }
}

// --- out = concat(x2[src], x2[dst]) (E x 16) @ WefP(16 x 16) + bef, WMMA ----
// kk in {0,1} reads the src half (k0<8), kk in {2,3} the dst half: uniform.
__global__ void k_final_wmma(const int* __restrict__ src, const int* __restrict__ dst,
                             const float* __restrict__ x2, const float* __restrict__ WefP,
                             const float* __restrict__ bef, float* __restrict__ out,
                             int E_, int nTiles) {
  int wave = (blockIdx.x * blockDim.x + threadIdx.x) >> 5;
  int lane = threadIdx.x & 31;
  if (wave >= nTiles) return;            // wave-uniform exit
  int half = lane >> 4;
  int m = lane & 15;
  int eRow = wave * 16 + m;
  int eC = eRow < E_ ? eRow : E_ - 1;    // clamp: garbage rows are store-guarded
  int sN = src[eC];
  int dN = dst[eC];
  v2f af[4], bf[4];
#pragma unroll
  for (int kk = 0; kk < 4; ++kk) {
    int k0 = kk * 4 + 2 * half;          // even; {k0,k0+1} never straddle 8
    const float* rp = (kk < 2) ? (x2 + sN * 8 + k0) : (x2 + dN * 8 + (k0 - 8));
    af[kk] = *(const v2f*)rp;
    bf[kk].x = WefP[k0 * 16 + m];
    bf[kk].y = WefP[(k0 + 1) * 16 + m];
  }
  v8f acc = {};
#pragma unroll
  for (int kk = 0; kk < 4; ++kk)
    acc = __builtin_amdgcn_wmma_f32_16x16x4_f32(false, af[kk], false, bf[kk],
                                                (short)0, acc, false, false);
  if (m < 8) {
    float bias = bef[m];
#pragma unroll
    for (int v = 0; v < 8; ++v) {
      int row = wave * 16 + v + 8 * half;
      if (row < E_) out[row * 8 + m] = acc[v] + bias;
    }
  }
}

// ---------------------------------------------------------------------------
extern "C" void kernel_launch(void* const* d_in, const int* in_sizes, int n_in,
                              void* d_out, int out_size, void* d_ws, size_t ws_size,
                              hipStream_t stream) {
  const int* src        = (const int*)d_in[0];
  const int* dst        = (const int*)d_in[1];
  const float* efeat    = (const float*)d_in[2];
  const float* capacity = (const float*)d_in[3];
  const float* W1  = (const float*)d_in[4];
  const float* We1 = (const float*)d_in[5];
  const float* al1 = (const float*)d_in[6];
  const float* ar1 = (const float*)d_in[7];
  const float* ae1 = (const float*)d_in[8];
  const float* b1  = (const float*)d_in[9];
  const float* W2  = (const float*)d_in[10];
  const float* We2 = (const float*)d_in[11];
  const float* al2 = (const float*)d_in[12];
  const float* ar2 = (const float*)d_in[13];
  const float* ae2 = (const float*)d_in[14];
  const float* b2  = (const float*)d_in[15];
  const float* Wef = (const float*)d_in[16];
  const float* bef = (const float*)d_in[17];
  float* out = (float*)d_out;

  const int E  = in_sizes[0];     // 400000
  const int Et = E + NN;          // +self loops

  // workspace carve-up (fp32 units)
  float* ws = (float*)d_ws;
  size_t o = 0;
  float*    coef  = ws + o; o += 384;            // coeffs + padded Wef
  float*    nfeat = ws + o; o += NN;
  unsigned* m1    = (unsigned*)(ws + o); o += (size_t)NN * 16;
  float*    den1  = ws + o; o += (size_t)NN * 16;
  float*    P1    = ws + o; o += (size_t)NN * 16;
  float*    Q1    = ws + o; o += (size_t)NN * 16;
  float*    x1    = ws + o; o += (size_t)NN * 16;
  float*    feat2 = ws + o; o += (size_t)NN * 64;
  float*    el2   = ws + o; o += (size_t)NN * 8;
  float*    er2   = ws + o; o += (size_t)NN * 8;
  unsigned* m2    = (unsigned*)(ws + o); o += (size_t)NN * 8;
  float*    den2  = ws + o; o += (size_t)NN * 8;
  float*    x2    = ws + o; o += (size_t)NN * 8;
  (void)ws_size; (void)n_in; (void)out_size;

  const int T = 256;
  auto nb = [](long n, int t) { return (int)((n + t - 1) / t); };

  // coefficients + padded Wef
  k_prep<<<1, 256, 0, stream>>>(W1, We1, al1, ar1, ae1, b1, We2, ae2, b2, Wef, coef);

  // layer 1
  k_init1<<<nb((long)NN * 16, T), T, 0, stream>>>(nfeat, m1, den1, P1, Q1, NN * 16);
  k_nfeat<<<nb(E, T), T, 0, stream>>>(dst, capacity, nfeat, E);
  k_passA1<<<nb(Et, T), T, 0, stream>>>(src, dst, efeat, nfeat, coef, m1, E, Et);
  k_passB1<<<nb(Et, T), T, 0, stream>>>(src, dst, efeat, nfeat, coef, m1, den1, E, Et);
  k_passC1<<<nb(Et, T), T, 0, stream>>>(src, dst, efeat, nfeat, coef, m1, den1, P1, Q1, E, Et);
  k_x1<<<nb((long)NN * 16, T), T, 0, stream>>>(W1, We1, P1, Q1, coef, x1, NN * 16);

  // layer 2 dense projection (WMMA)
  int wavesF2 = ((NN + 15) / 16) * 4;
  k_feat2_wmma<<<nb((long)wavesF2 * 32, T), T, 0, stream>>>(x1, W2, feat2, wavesF2);

  k_el2er2<<<nb((long)NN * 8, T), T, 0, stream>>>(feat2, al2, ar2, el2, er2, NN * 8);
  k_init2<<<nb((long)NN * 8, T), T, 0, stream>>>(m2, den2, x2, coef, NN * 8);
  k_passA2<<<nb(Et, T), T, 0, stream>>>(src, dst, efeat, el2, er2, coef, m2, E, Et);
  k_passB2<<<nb(Et, T), T, 0, stream>>>(src, dst, efeat, el2, er2, coef, m2, den2, E, Et);
  k_passC2<<<nb(Et, T), T, 0, stream>>>(src, dst, efeat, el2, er2, coef, m2, den2,
                                        feat2, We2, x2, E, Et);

  // final linear head (WMMA)
  int tiles = (E + 15) / 16;
  k_final_wmma<<<nb((long)tiles * 32, T), T, 0, stream>>>(src, dst, x2,
                                                          coef + 128, bef, out,
                                                          E, tiles);
}